// BruteForce_1486058685043
// MI455X (gfx1250) — compile-verified
//
#include <hip/hip_runtime.h>
#include <stdint.h>

typedef __attribute__((ext_vector_type(2))) float v2f;
typedef __attribute__((ext_vector_type(8))) float v8f;

#define KTOP 100
#define CHUNK 2048
#define GPB 4            // query-groups (of 16) per block in phase 1

// map float -> monotonically ordered u32 (larger float => larger u32)
__device__ __forceinline__ uint32_t ordf(float f) {
    uint32_t b = __float_as_uint(f);
    return b ^ (0x80000000u | (uint32_t)((int32_t)b >> 31));
}

// Executed by one wave. hist: nrep replicated 256-bin histograms, stride256 u32 apart.
// Scans bins 255..0 for the bin where remaining rank *s_kk lands; updates *s_pref/*s_kk.
__device__ __forceinline__ void pick_bin(const uint32_t* hist, int nrep, int stride256,
                                         uint32_t* s_pref, uint32_t* s_kk,
                                         int lane, int shift) {
    uint32_t c[8];
    uint32_t lsum = 0;
    #pragma unroll
    for (int j = 0; j < 8; ++j) {
        int bin = 255 - (lane * 8 + j);
        uint32_t v = 0;
        #pragma unroll
        for (int wv = 0; wv < 8; ++wv)
            v += (wv < nrep) ? hist[wv * stride256 + bin] : 0u;
        c[j] = v; lsum += v;
    }
    uint32_t incl = lsum;
    #pragma unroll
    for (int ofs = 1; ofs < 32; ofs <<= 1) {
        uint32_t o = __shfl_up(incl, ofs);
        if (lane >= ofs) incl += o;
    }
    uint32_t excl = incl - lsum;
    uint32_t kk = *s_kk;
    if (excl < kk && kk <= incl) {
        uint32_t run = excl;
        #pragma unroll
        for (int j = 0; j < 8; ++j) {
            uint32_t nr = run + c[j];
            if (run < kk && kk <= nr) {
                *s_pref |= ((uint32_t)(255 - (lane * 8 + j))) << shift;
                *s_kk = kk - run;
            }
            run = nr;
        }
    }
}

// ---------------- Phase 1: WMMA scoring + per-(query,chunk) exact top-100 ----------------
__global__ __launch_bounds__(256, 1)
void topk_phase1(const float* __restrict__ Q, const float* __restrict__ Cand,
                 int N, int nchunk,
                 uint32_t* __restrict__ wsScore, int* __restrict__ wsId) {
    __shared__ uint32_t sc[16][CHUNK];              // 128 KB: ordered scores, row=query
    __shared__ uint32_t hist[8][256];               // 8 KB: per-wave replicated histograms
    __shared__ uint32_t s_pref[2], s_kk[2], s_cgt[2], s_ceq[2];

    const int tid   = threadIdx.x;
    const int wave  = tid >> 5;
    const int lane  = tid & 31;
    const int n16   = lane & 15;
    const int hhalf = lane >> 4;
    const int chunk = blockIdx.x;
    const int cbase = chunk * CHUNK + wave * 256;   // this wave's candidate base
    const int tsub  = tid & 127;                    // selection: lane within query team
    const int qsel  = tid >> 7;                     // selection: 0 -> query r, 1 -> query r+8

    #pragma unroll 1
    for (int gi = 0; gi < GPB; ++gi) {
        const int g = blockIdx.y * GPB + gi;        // query group (16 queries)

        // A fragments: queries, 16x4 f32 per k-step. ~32 VGPRs, persistent.
        v2f qa[16];
        const float* qrow = Q + (size_t)(g * 16 + n16) * 64 + 2 * hhalf;
        #pragma unroll
        for (int kk = 0; kk < 16; ++kk) qa[kk] = *(const v2f*)(qrow + 4 * kk);

        // 16 WMMA tiles of 16 candidates each (256 candidates per wave), rolled.
        #pragma unroll 1
        for (int t = 0; t < 16; ++t) {
            int cand = cbase + t * 16 + n16;
            int crow = cand < N ? cand : N - 1;     // clamp tail (masked below)
            const float* cptr = Cand + (size_t)crow * 64 + 2 * hhalf;
            __builtin_prefetch((const void*)(cptr + 16 * 64), 0, 1);
            v8f a = {0.f,0.f,0.f,0.f,0.f,0.f,0.f,0.f};
            #pragma unroll
            for (int kk = 0; kk < 16; ++kk) {
                v2f b = *(const v2f*)(cptr + 4 * kk);
                a = __builtin_amdgcn_wmma_f32_16x16x4_f32(
                        false, qa[kk], false, b, (short)0, a, false, false);
            }
            // Store 8 rows (queries r+8*hhalf) for this candidate column into LDS.
            const int col = wave * 256 + t * 16 + n16;
            const bool valid = cand < N;
            #pragma unroll
            for (int r = 0; r < 8; ++r)
                sc[r + 8 * hhalf][col] = valid ? ordf(a[r]) : 0u;
        }
        __syncthreads();

        // Selection: 8 rounds; round r -> queries (g*16+r) by waves 0-3 and
        // (g*16+r+8) by waves 4-7, each team scanning its 2048 LDS scores.
        #pragma unroll 1
        for (int r = 0; r < 8; ++r) {
            const uint32_t* row = &sc[r + 8 * qsel][0];
            if (tid < 2) { s_pref[tid] = 0u; s_kk[tid] = KTOP; s_cgt[tid] = 0u; s_ceq[tid] = 0u; }
            __syncthreads();
            #pragma unroll 1
            for (int d = 3; d >= 0; --d) {
                for (int b = lane; b < 256; b += 32) hist[wave][b] = 0u;
                __syncthreads();
                const uint32_t prefMask = (d == 3) ? 0u : (0xFFFFFFFFu << ((d + 1) * 8));
                const uint32_t prefv = s_pref[qsel];
                const int shift = d * 8;
                #pragma unroll
                for (int i = 0; i < 16; ++i) {
                    uint32_t u = row[tsub + 128 * i];
                    if ((u & prefMask) == prefv)
                        atomicAdd(&hist[wave][(u >> shift) & 255], 1u);
                }
                __syncthreads();
                if (wave == 0) pick_bin(&hist[0][0], 4, 256, &s_pref[0], &s_kk[0], lane, shift);
                if (wave == 4) pick_bin(&hist[4][0], 4, 256, &s_pref[1], &s_kk[1], lane, shift);
                __syncthreads();
            }
            // Compact exactly KTOP entries for this (query, chunk).
            const uint32_t T    = s_pref[qsel];
            const uint32_t need = s_kk[qsel];
            const uint32_t ngt  = KTOP - need;
            const int q = g * 16 + r + 8 * qsel;
            const size_t obase = ((size_t)q * nchunk + chunk) * KTOP;
            #pragma unroll
            for (int i = 0; i < 16; ++i) {
                const int col = tsub + 128 * i;
                uint32_t u = row[col];
                int slot = -1;
                if (u > T) {
                    slot = (int)atomicAdd(&s_cgt[qsel], 1u);
                } else if (u == T) {
                    uint32_t e = atomicAdd(&s_ceq[qsel], 1u);
                    if (e < need) slot = (int)(ngt + e);
                }
                if (slot >= 0) { wsScore[obase + slot] = u; wsId[obase + slot] = chunk * CHUNK + col; }
            }
            __syncthreads();
        }
    }
}

// ---------------- Phase 2: merge partials -> exact global top-100, sorted ----------------
__global__ __launch_bounds__(256, 1)
void topk_phase2(const uint32_t* __restrict__ wsScore, const int* __restrict__ wsId,
                 const int* __restrict__ idmap, int npart,
                 float* __restrict__ outScore, int* __restrict__ outId) {
    __shared__ uint32_t hist[8][256];
    __shared__ uint32_t s_pref, s_kk, s_cgt, s_ceq;
    __shared__ unsigned long long keybuf[128];

    const int tid  = threadIdx.x;
    const int wave = tid >> 5;
    const int lane = tid & 31;
    const int q = blockIdx.x;
    const uint32_t* scp = wsScore + (size_t)q * npart;
    const int* idp = wsId + (size_t)q * npart;

    if (tid == 0) { s_pref = 0u; s_kk = KTOP; s_cgt = 0u; s_ceq = 0u; }
    if (tid < 128) keybuf[tid] = 0ull;
    __syncthreads();

    #pragma unroll 1
    for (int d = 3; d >= 0; --d) {
        for (int b = lane; b < 256; b += 32) hist[wave][b] = 0u;
        __syncthreads();
        const uint32_t prefMask = (d == 3) ? 0u : (0xFFFFFFFFu << ((d + 1) * 8));
        const uint32_t prefv = s_pref;
        const int shift = d * 8;
        for (int i = tid; i < npart; i += 256) {
            uint32_t u = scp[i];
            if ((u & prefMask) == prefv) atomicAdd(&hist[wave][(u >> shift) & 255], 1u);
        }
        __syncthreads();
        if (wave == 0) pick_bin(&hist[0][0], 8, 256, &s_pref, &s_kk, lane, shift);
        __syncthreads();
    }

    const uint32_t T    = s_pref;
    const uint32_t need = s_kk;
    const uint32_t ngt  = KTOP - need;
    for (int i = tid; i < npart; i += 256) {
        uint32_t u = scp[i];
        int slot = -1;
        if (u > T) {
            slot = (int)atomicAdd(&s_cgt, 1u);
        } else if (u == T) {
            uint32_t e = atomicAdd(&s_ceq, 1u);
            if (e < need) slot = (int)(ngt + e);
        }
        if (slot >= 0)
            keybuf[slot] = ((unsigned long long)u << 32) | (uint32_t)(~(uint32_t)idp[i]);
    }
    __syncthreads();

    // Bitonic sort 128 keys, descending (ties: smaller id first via ~id in low bits).
    #pragma unroll 1
    for (int ksz = 2; ksz <= 128; ksz <<= 1) {
        #pragma unroll 1
        for (int jst = ksz >> 1; jst > 0; jst >>= 1) {
            if (tid < 128) {
                int ixj = tid ^ jst;
                if (ixj > tid) {
                    unsigned long long a = keybuf[tid], b = keybuf[ixj];
                    bool up = ((tid & ksz) == 0);
                    if (up ? (a < b) : (a > b)) { keybuf[tid] = b; keybuf[ixj] = a; }
                }
            }
            __syncthreads();
        }
    }

    if (tid < KTOP) {
        unsigned long long k = keybuf[tid];
        uint32_t u  = (uint32_t)(k >> 32);
        uint32_t fb = (u & 0x80000000u) ? (u ^ 0x80000000u) : ~u;  // inverse of ordf
        int cidx = (int)(~(uint32_t)k);
        outScore[(size_t)q * KTOP + tid] = __uint_as_float(fb);
        outId[(size_t)q * KTOP + tid] = (cidx >= 0) ? idmap[cidx] : -1;
    }
}

extern "C" void kernel_launch(void* const* d_in, const int* in_sizes, int n_in,
                              void* d_out, int out_size, void* d_ws, size_t ws_size,
                              hipStream_t stream) {
    const float* Q    = (const float*)d_in[0];   // [B, 64] f32
    const float* Cand = (const float*)d_in[1];   // [N, 64] f32
    const int*   ids  = (const int*)d_in[2];     // [N] i32

    const int Bq = in_sizes[0] / 64;             // 256
    const int N  = in_sizes[2];                  // 1,000,000
    const int nchunk = (N + CHUNK - 1) / CHUNK;
    const int npart  = nchunk * KTOP;

    uint32_t* wsScore = (uint32_t*)d_ws;                        // [B, nchunk*KTOP] u32
    int*      wsId    = (int*)(wsScore + (size_t)Bq * npart);   // [B, nchunk*KTOP] i32

    dim3 grid1((unsigned)nchunk, (unsigned)(Bq / (16 * GPB)));
    topk_phase1<<<grid1, 256, 0, stream>>>(Q, Cand, N, nchunk, wsScore, wsId);

    float* outScore = (float*)d_out;                     // [B, KTOP] f32
    int*   outId    = (int*)d_out + (size_t)Bq * KTOP;   // [B, KTOP] i32
    topk_phase2<<<Bq, 256, 0, stream>>>(wsScore, wsId, ids, npart, outScore, outId);
}